// SumLayer_34823594836341
// MI455X (gfx1250) — compile-verified
//
#include <hip/hip_runtime.h>

#define C_DIM 64
#define B_DIM 256

typedef int v4i __attribute__((ext_vector_type(4)));
typedef __attribute__((address_space(1))) v4i gv4i;   // global int4
typedef __attribute__((address_space(3))) v4i lv4i;   // LDS int4

// ---- CDNA5 async global->LDS copy (tracked by ASYNCcnt) -------------------
__device__ __forceinline__ void async_copy_b128(const float* gsrc, float* lds_dst) {
#if __has_builtin(__builtin_amdgcn_global_load_async_to_lds_b128)
    __builtin_amdgcn_global_load_async_to_lds_b128(
        (gv4i*)gsrc, (lv4i*)lds_dst, /*offset=*/0, /*cpol=*/0);
#else
    *(float4*)lds_dst = *(const float4*)gsrc;   // sync fallback
#endif
}

__device__ __forceinline__ void wait_async_all() {
#if __has_builtin(__builtin_amdgcn_s_wait_asynccnt)
    __builtin_amdgcn_s_wait_asynccnt(0);
#else
    asm volatile("s_wait_asynccnt 0x0" ::: "memory");
#endif
}

// ---- fused gather + weighted logsumexp ------------------------------------
// One block per node group: stage the 64x256 f32 child tile into LDS with
// async B128 copies (ASYNCcnt), then one batch column per thread. Weights are
// staged once into LDS and read via conflict-free broadcasts.
__global__ __launch_bounds__(256) void sum_layer_kernel(
    const float* __restrict__ element_mars,
    const float* __restrict__ params,
    const int*   __restrict__ nids,
    const int*   __restrict__ cids,
    const int*   __restrict__ pids,
    float*       __restrict__ out) {
    __shared__ float tile[C_DIM * B_DIM];   // 64 KB (4 blocks fit in 320 KB WGP LDS)
    __shared__ float wsh[C_DIM];
    __shared__ int   rows[C_DIM];

    const int n = blockIdx.x;
    const int t = threadIdx.x;

    if (t < C_DIM) {
        rows[t] = cids[n * C_DIM + t];
        wsh[t]  = params[pids[n * C_DIM + t]];
    }
    __syncthreads();

    // 256 lanes x 16B = 4 rows (1 KB each) per pass; 16 passes cover 64 rows.
    const int c0 = t >> 6;        // 0..3 : row within a pass
    const int q  = (t & 63) << 2; // float index of this lane's float4 in the row
    #pragma unroll
    for (int i = 0; i < 16; ++i) {
        const int c = (i << 2) + c0;
        const float* gsrc = element_mars + (size_t)rows[c] * B_DIM + q;
        async_copy_b128(gsrc, &tile[c * B_DIM + q]);
    }
    wait_async_all();
    __syncthreads();

    // Thread t owns batch column t; the 64 child values stay in VGPRs so the
    // LDS tile is read exactly once (stride-1 across lanes, and the compiler
    // pairs the c-strided reads into ds_load_2addr_stride64_b32).
    float x[C_DIM];
    float m = -3.402823466e38f;
    #pragma unroll
    for (int c = 0; c < C_DIM; ++c) {
        x[c] = tile[c * B_DIM + t];
        m = fmaxf(m, x[c]);
    }
    float s = 0.0f;
    #pragma unroll
    for (int c = 0; c < C_DIM; ++c) {
        s = __builtin_fmaf(__expf(x[c] - m), wsh[c], s);
    }
    const float r = __logf(fmaxf(s, 1e-10f)) + m;
    out[(size_t)nids[n] * B_DIM + t] = r;
}

// ---- copy node_mars -> d_out (rows not in nids must pass through) ---------
__global__ __launch_bounds__(256) void copy_f4_kernel(const float4* __restrict__ src,
                                                      float4* __restrict__ dst, int n4) {
    int i = blockIdx.x * blockDim.x + threadIdx.x;
    if (i < n4) dst[i] = src[i];
}

// ---- host-side launch -----------------------------------------------------
extern "C" void kernel_launch(void* const* d_in, const int* in_sizes, int n_in,
                              void* d_out, int out_size, void* d_ws, size_t ws_size,
                              hipStream_t stream) {
    const float* node_mars    = (const float*)d_in[0];
    const float* element_mars = (const float*)d_in[1];
    const float* params       = (const float*)d_in[2];
    const int*   nids         = (const int*)d_in[3];
    const int*   cids         = (const int*)d_in[4];
    const int*   pids         = (const int*)d_in[5];
    float*       out          = (float*)d_out;

    const int N = in_sizes[3];          // number of node groups (4096)

    // 1) pass-through copy of node_mars into d_out (8 MB, float4).
    const int n4 = out_size / 4;
    copy_f4_kernel<<<(n4 + 255) / 256, 256, 0, stream>>>(
        (const float4*)node_mars, (float4*)out, n4);

    // 2) fused gather + weighted logsumexp, overwriting rows nids[n].
    sum_layer_kernel<<<N, 256, 0, stream>>>(element_mars, params, nids, cids,
                                            pids, out);
}